// NodeSage_70368744177918
// MI455X (gfx1250) — compile-verified
//
#include <hip/hip_runtime.h>

// GraphSAGE 3-layer inference for MI455X (gfx1250, wave32).
// transform-then-aggregate; fp32 WMMA 16x16x4 GEMMs with LDS-staged,
// bank-conflict-free operand tiles; L2-resident f32 atomics for the scatter.

typedef __attribute__((ext_vector_type(2))) float v2f;
typedef __attribute__((ext_vector_type(8))) float v8f;

// ---------------------------------------------------------------- zero fill
__global__ void zero_f4_kernel(float4* __restrict__ p, long n4) {
  long i = (long)blockIdx.x * blockDim.x + threadIdx.x;
  if (i < n4) p[i] = make_float4(0.f, 0.f, 0.f, 0.f);
}

// ------------------------------------------------------- degree count / 1/x
__global__ void count_kernel(const long long* __restrict__ dst, int E,
                             float* __restrict__ cnt) {
  int e = blockIdx.x * blockDim.x + threadIdx.x;
  if (e < E) {
    int d = (int)dst[e];
    (void)__hip_atomic_fetch_add(&cnt[d], 1.0f, __ATOMIC_RELAXED,
                                 __HIP_MEMORY_SCOPE_AGENT);
  }
}

__global__ void recip_kernel(float* __restrict__ cnt, int N) {
  int n = blockIdx.x * blockDim.x + threadIdx.x;
  if (n < N) cnt[n] = 1.0f / fmaxf(cnt[n], 1.0f);
}

// ------------------------------------------------ OUT[N,J] = X[N,K] @ W[J,K]^T
// Block = 8 waves, 16 output rows, all J columns. Both operands staged in
// dynamic LDS with row stride PAD=132 floats (132 mod 64 banks = 4 -> each
// half-wave's ds_load_b64 hits 32 distinct banks; 16B alignment kept).
// A fragments preloaded into registers; then an uninterrupted chain of
// K/4 fp32 WMMAs per 16x16 tile.
template <int K, int J>
__launch_bounds__(256)
__global__ void gemm_xwt_kernel(const float* __restrict__ X,
                                const float* __restrict__ W,   // [J, K]
                                float* __restrict__ OUT,       // [N, J]
                                int N) {
  constexpr int PAD = 132;            // LDS row stride (floats), conflict-free
  constexpr int NK4 = K / 4;          // WMMA k-steps
  extern __shared__ float smem[];
  float* Xs = smem;                   // [16][PAD]
  float* Ws = smem + 16 * PAD;        // [J][PAD]

  const int m0 = blockIdx.x * 16;

  // stage W (J x K) into padded LDS, coalesced float4
  for (int idx = threadIdx.x; idx < J * NK4; idx += 256) {
    const int r = idx / NK4, c = (idx % NK4) * 4;
    *(float4*)(Ws + r * PAD + c) = *(const float4*)(W + (long)r * K + c);
  }
  // stage X tile (16 x K), zero-fill rows past N (uniform guard, no per-k exec)
  for (int idx = threadIdx.x; idx < 16 * NK4; idx += 256) {
    const int r = idx / NK4, c = (idx % NK4) * 4;
    float4 v = make_float4(0.f, 0.f, 0.f, 0.f);
    if (m0 + r < N) v = *(const float4*)(X + (long)(m0 + r) * K + c);
    *(float4*)(Xs + r * PAD + c) = v;
  }
  __syncthreads();

  const int lane = threadIdx.x & 31;
  const int wave = threadIdx.x >> 5;
  const int hr   = lane & 15;         // fragment row within 16-row tile
  const int koff = (lane >> 4) * 2;   // lanes 0-15: K={k,k+1}, 16-31: K={k+2,k+3}

  // preload all A fragments (reused across n-tiles)
  const float* xsrow = Xs + hr * PAD + koff;
  v2f a[NK4];
#pragma unroll
  for (int k = 0; k < NK4; ++k) {
    float2 t = *(const float2*)(xsrow + k * 4);
    a[k].x = t.x; a[k].y = t.y;
  }

  for (int nt = wave; nt < J / 16; nt += 8) {
    const int n0 = nt * 16;
    const float* wsrow = Ws + (n0 + hr) * PAD + koff;
    v8f acc = {};
#pragma unroll
    for (int k = 0; k < NK4; ++k) {
      float2 t = *(const float2*)(wsrow + k * 4);
      v2f b; b.x = t.x; b.y = t.y;
      acc = __builtin_amdgcn_wmma_f32_16x16x4_f32(
          false, a[k], false, b, (short)0, acc, false, false);
    }
    // C/D layout: VGPR v -> row m0 + v + (lane>=16)*8, col n0 + (lane&15)
    const int rbase = m0 + (lane >> 4) * 8;
    const int col   = n0 + hr;
#pragma unroll
    for (int v = 0; v < 8; ++v) {
      const int r = rbase + v;
      if (r < N) OUT[(long)r * J + col] = acc[v];
    }
  }
}

// ------------------------------------- edge scatter: agg[dst] += XL[src]
// One wave per edge; indices forced scalar (s_load) via readfirstlane;
// each lane owns D/32 consecutive floats: coalesced 512B gather per wave,
// then non-returning f32 atomics that resolve in L2 (agg fits in 192MB L2).
template <int D>
__global__ void scatter_kernel(const float* __restrict__ XL,
                               const long long* __restrict__ src,
                               const long long* __restrict__ dst,
                               float* __restrict__ agg, int E) {
  const int lane = threadIdx.x & 31;
  int ew = (int)(((long)blockIdx.x * blockDim.x + threadIdx.x) >> 5);
  ew = __builtin_amdgcn_readfirstlane(ew);   // wave-uniform by construction
  if (ew >= E) return;
  const int s = (int)src[ew];
  const int d = (int)dst[ew];
  constexpr int V = D / 32;                  // 4 (d=128) or 2 (d=64)
  const float* xp = XL + (long)s * D + lane * V;
  float*       ap = agg + (long)d * D + lane * V;
  float vals[V];
  if constexpr (V == 4) {
    float4 t = *(const float4*)xp;
    vals[0] = t.x; vals[1] = t.y; vals[2] = t.z; vals[3] = t.w;
  } else {
    float2 t = *(const float2*)xp;
    vals[0] = t.x; vals[1] = t.y;
  }
#pragma unroll
  for (int i = 0; i < V; ++i)
    (void)__hip_atomic_fetch_add(&ap[i], vals[i], __ATOMIC_RELAXED,
                                 __HIP_MEMORY_SCOPE_AGENT);
}

// ------------------------- epilogue: out = [relu](agg*inv[n] + xr + bias[j])
template <bool RELU, int J>
__global__ void combine_kernel(const float* __restrict__ agg,
                               const float* __restrict__ xr,
                               const float* __restrict__ bias,
                               const float* __restrict__ inv,
                               float* __restrict__ out, int N) {
  long i4 = (long)blockIdx.x * blockDim.x + threadIdx.x;
  long total4 = (long)N * J / 4;
  if (i4 >= total4) return;
  long i = i4 * 4;
  int n = (int)(i / J);
  int j = (int)(i % J);
  float4 a = *(const float4*)(agg + i);
  float4 r = *(const float4*)(xr + i);
  float4 b = *(const float4*)(bias + j);
  float  s = inv[n];
  float4 o;
  o.x = fmaf(a.x, s, r.x + b.x);
  o.y = fmaf(a.y, s, r.y + b.y);
  o.z = fmaf(a.z, s, r.z + b.z);
  o.w = fmaf(a.w, s, r.w + b.w);
  if (RELU) {
    o.x = fmaxf(o.x, 0.f); o.y = fmaxf(o.y, 0.f);
    o.z = fmaxf(o.z, 0.f); o.w = fmaxf(o.w, 0.f);
  }
  *(float4*)(out + i) = o;
}

// --------------------------------------------------------------- launcher
extern "C" void kernel_launch(void* const* d_in, const int* in_sizes, int n_in,
                              void* d_out, int out_size, void* d_ws, size_t ws_size,
                              hipStream_t stream) {
  const float* x   = (const float*)d_in[0];
  const float* Wl0 = (const float*)d_in[1];
  const float* Wr0 = (const float*)d_in[2];
  const float* b0  = (const float*)d_in[3];
  const float* Wl1 = (const float*)d_in[4];
  const float* Wr1 = (const float*)d_in[5];
  const float* b1  = (const float*)d_in[6];
  const float* Wl2 = (const float*)d_in[7];
  const float* Wr2 = (const float*)d_in[8];
  const float* b2  = (const float*)d_in[9];
  const long long* edge = (const long long*)d_in[10];  // int64 [2, E]

  const int N = in_sizes[0] / 128;
  const int E = in_sizes[10] / 2;
  const long long* srcI = edge;
  const long long* dstI = edge + E;

  // workspace layout (floats): h | xl | xr | agg | inv
  float* h   = (float*)d_ws;
  float* xl  = h   + (long)N * 128;
  float* xr  = xl  + (long)N * 128;
  float* agg = xr  + (long)N * 128;
  float* inv = agg + (long)N * 128;
  float* out = (float*)d_out;

  const int mGrid   = (N + 15) / 16;           // 16-row GEMM tiles
  const int eBlocks = (E + 7) / 8;             // 8 edges (waves) per 256-thr block
  const long n4_128 = (long)N * 128 / 4;
  const long n4_64  = (long)N * 64 / 4;
  const int  zGrid128 = (int)((n4_128 + 255) / 256);
  const int  zGrid64  = (int)((n4_64  + 255) / 256);

  constexpr int PAD = 132;
  const size_t lds128 = (size_t)(16 + 128) * PAD * sizeof(float);  // ~76 KB
  const size_t lds64  = (size_t)(16 + 64)  * PAD * sizeof(float);  // ~42 KB

  // ---- inverse degrees (shared by all layers) ----
  {
    long n4 = (N + 3) / 4;
    zero_f4_kernel<<<(int)((n4 + 255) / 256), 256, 0, stream>>>((float4*)inv, n4);
    count_kernel<<<(E + 255) / 256, 256, 0, stream>>>(dstI, E, inv);
    recip_kernel<<<(N + 255) / 256, 256, 0, stream>>>(inv, N);
  }

  // ---- layer 0: x(128) -> h(128), relu ----
  gemm_xwt_kernel<128, 128><<<mGrid, 256, lds128, stream>>>(x, Wl0, xl, N);
  gemm_xwt_kernel<128, 128><<<mGrid, 256, lds128, stream>>>(x, Wr0, xr, N);
  zero_f4_kernel<<<zGrid128, 256, 0, stream>>>((float4*)agg, n4_128);
  scatter_kernel<128><<<eBlocks, 256, 0, stream>>>(xl, srcI, dstI, agg, E);
  combine_kernel<true, 128><<<zGrid128, 256, 0, stream>>>(agg, xr, b0, inv, h, N);

  // ---- layer 1: h(128) -> h(128), relu ----
  gemm_xwt_kernel<128, 128><<<mGrid, 256, lds128, stream>>>(h, Wl1, xl, N);
  gemm_xwt_kernel<128, 128><<<mGrid, 256, lds128, stream>>>(h, Wr1, xr, N);
  zero_f4_kernel<<<zGrid128, 256, 0, stream>>>((float4*)agg, n4_128);
  scatter_kernel<128><<<eBlocks, 256, 0, stream>>>(xl, srcI, dstI, agg, E);
  combine_kernel<true, 128><<<zGrid128, 256, 0, stream>>>(agg, xr, b1, inv, h, N);

  // ---- layer 2: h(128) -> out(64), no relu ----
  gemm_xwt_kernel<128, 64><<<mGrid, 256, lds64, stream>>>(h, Wl2, xl, N);
  gemm_xwt_kernel<128, 64><<<mGrid, 256, lds64, stream>>>(h, Wr2, xr, N);
  zero_f4_kernel<<<zGrid64, 256, 0, stream>>>((float4*)agg, n4_64);
  scatter_kernel<64><<<eBlocks, 256, 0, stream>>>(xl, srcI, dstI, agg, E);
  combine_kernel<false, 64><<<zGrid64, 256, 0, stream>>>(agg, xr, b2, inv, out, N);
}